// ChannelCapacity1Inf_90237262889791
// MI455X (gfx1250) — compile-verified
//
#include <hip/hip_runtime.h>
#include <hip/hip_bf16.h>

typedef __attribute__((ext_vector_type(2))) float v2f;
typedef __attribute__((ext_vector_type(8))) float v8f;

#define BLOCK 256
#define WAVES (BLOCK / 32)

// Gauss-Legendre order-8 nodes/weights mapped to [0,1] (matches reference)
#define GL_NODES_F  {0.01985507175123186f, 0.10166676129318664f, 0.23723379504183550f, \
                     0.40828267875217510f, 0.59171732124782490f, 0.76276620495816450f, \
                     0.89833323870681340f, 0.98014492824876820f}
#define GL_WTS_F    {0.05061426814518813f, 0.11119051722668723f, 0.15685332293894365f, \
                     0.18134188834145102f, 0.18134188834145102f, 0.15685332293894365f, \
                     0.11119051722668723f, 0.05061426814518813f}

// f(lam): eigenvalue map of the reference psd_logm (6 sqrtms + GL8 Pade of log(1+x)).
// Raw v_sqrt_f32 / v_rcp_f32: denominators 1+t*x >= 1-t > 0.0198, well conditioned.
__device__ __forceinline__ float logm_eig_f32(float lam) {
    float r = lam;
#pragma unroll
    for (int i = 0; i < 6; ++i) r = __builtin_amdgcn_sqrtf(r);
    float x = r - 1.0f;
    const float tn[8] = GL_NODES_F;
    const float wn[8] = GL_WTS_F;
    float s = 0.0f;
#pragma unroll
    for (int i = 0; i < 8; ++i)
        s = fmaf(wn[i] * x, __builtin_amdgcn_rcpf(fmaf(tn[i], x, 1.0f)), s);
    return 64.0f * s;
}

__device__ __forceinline__ double logm_eig_f64(double lam) {
    double r = lam;
#pragma unroll
    for (int i = 0; i < 6; ++i) r = sqrt(r);
    double x = r - 1.0;
    const double tn[8] = {0.01985507175123186, 0.10166676129318664, 0.2372337950418355,
                          0.4082826787521751, 0.5917173212478249, 0.7627662049581645,
                          0.8983332387068133, 0.9801449282487682};
    const double wn[8] = {0.05061426814518813, 0.11119051722668723, 0.15685332293894365,
                          0.18134188834145102, 0.18134188834145102, 0.15685332293894365,
                          0.11119051722668723, 0.05061426814518813};
    double s = 0.0;
#pragma unroll
    for (int i = 0; i < 8; ++i) s += wn[i] * x / fma(tn[i], x, 1.0);
    return 64.0 * s;
}

// Compiler-only scheduling fence: LDS ops from one wave are executed in order
// by the LDS pipeline, so wave-private store->load->store sequences need no
// hardware barrier, only protection from compiler reordering.
#define LDS_ORDER() asm volatile("" ::: "memory")

// Kernel 1: stream all states; f32 per-state math (hardware transcendentals),
// f64 per-thread accumulation of T = sum(t), TS = sum(t * S(sigma_n));
// WMMA f32 16x16x4 rank-4 updates accumulate M = V^T V for the mixture rho.
__global__ __launch_bounds__(BLOCK) void per_state_kernel(
    const double* __restrict__ theta_coeff,
    const double* __restrict__ theta_state,
    const double* __restrict__ kop,
    double* __restrict__ partials,
    int N, int total_threads, int iters)
{
    __shared__ float  stage[BLOCK][4];   // per-lane scaled state 4-vectors (wave-private slices)
    __shared__ float  wred[WAVES][6];    // per-wave extracted M combos
    __shared__ double dredT[BLOCK];
    __shared__ double dredS[BLOCK];

    const int tid  = threadIdx.x;
    const int lane = tid & 31;
    const int wave = tid >> 5;
    const int half = lane >> 4;   // A-matrix: lanes 16-31 hold K = 2,3
    const int row  = lane & 15;   // A-matrix: M index
    const int gtid = blockIdx.x * BLOCK + tid;

    const double sq1mg = kop[3];           // sqrt(1-g)
    const double sqg   = kop[5];           // sqrt(g)
    const float four_g_1mg = (float)(4.0 * (sqg * sqg) * (sq1mg * sq1mg));

    double accT = 0.0, accTS = 0.0;
    v8f cacc = {};                          // M = sum v v^T accumulator (f32)

    for (int it = 0; it < iters; ++it) {   // uniform trip count: EXEC all-ones at WMMA
        int n = gtid + it * total_threads;
        bool pred = (n < N);

        float t = 0.0f, x0 = 0.0f, x1 = 0.0f, x2 = 0.0f, x3 = 0.0f, inv = 0.0f;
        if (pred) {
            // pull next iteration's lines toward the WGP while this one computes
            int nn = n + total_threads;
            if (nn < N) {
                __builtin_prefetch(theta_state + 4ll * nn, 0, 3);
                __builtin_prefetch(theta_coeff + nn, 0, 3);
            }
            // softplus in f32 via hw exp2/log2: theta in (-1,1), no cancellation
            float tc = (float)theta_coeff[n];
            t = 0.6931471805599453f *
                __builtin_amdgcn_logf(1.0f + __builtin_amdgcn_exp2f(tc * 1.4426950408889634f));
            const double2* sp = (const double2*)(theta_state + 4ll * n);  // 2x b128
            double2 ab = sp[0], cd = sp[1];
            x0 = (float)ab.x; x1 = (float)ab.y; x2 = (float)cd.x; x3 = (float)cd.y;
            float nrm2 = fmaf(x0, x0, fmaf(x1, x1, fmaf(x2, x2, x3 * x3)));
            inv = __builtin_amdgcn_rcpf(nrm2);
        }

        // sigma_n eigenvalues in closed form: tr=1, det = g(1-g) p^2, p = |s1|^2
        float p    = fmaf(x2, x2, x3 * x3) * inv;
        float disc = __builtin_amdgcn_sqrtf(fmaxf(0.0f, fmaf(-four_g_1mg * p, p, 1.0f)));
        float lp = 0.5f * (1.0f + disc), lm = 0.5f * (1.0f - disc);
        float S  = -(lp * logm_eig_f32(lp) + lm * logm_eig_f32(lm));
        accT  += (double)t;
        accTS += (double)t * (double)S;

        // stage v = sqrt(t)/||theta|| * (Re0, Im0, Re1, Im1) for the rank-k update
        float vs = __builtin_amdgcn_sqrtf(t * inv);    // 0 when !pred
        stage[tid][0] = vs * x0;
        stage[tid][1] = vs * x1;
        stage[tid][2] = vs * x2;
        stage[tid][3] = vs * x3;
        LDS_ORDER();   // wave-private region: in-order DS pipeline, no block barrier

        // 8 WMMAs consume this wave's 32 states, 4 states (K=4) per op.
        // A[m,k] = v_k[m], B[k,n] = v_k[n]  ->  D += sum_k v_k v_k^T = M
#pragma unroll
        for (int q = 0; q < 8; ++q) {
            int kbase = (q << 2) + (half << 1);
            float a0 = 0.0f, a1 = 0.0f;
            if (row < 4) {
                a0 = stage[(wave << 5) + kbase + 0][row];
                a1 = stage[(wave << 5) + kbase + 1][row];
            }
            v2f av; av.x = a0; av.y = a1;
            cacc = __builtin_amdgcn_wmma_f32_16x16x4_f32(
                false, av, false, av, (short)0, cacc, false, false);
        }
        LDS_ORDER();   // keep next iteration's stores behind this iteration's loads
    }

    // Extract needed entries of M from the 16x16 f32 D layout (lane = N, vgpr = M):
    // U11 = M22+M33 ; Ure = M02+M13 ; Uim = M12-M03
    if (lane == 2) { wred[wave][0] = cacc[0]; wred[wave][1] = cacc[1]; wred[wave][2] = cacc[2]; }
    if (lane == 3) { wred[wave][3] = cacc[1]; wred[wave][4] = -cacc[0]; wred[wave][5] = cacc[3]; }
    dredT[tid] = accT;
    dredS[tid] = accTS;
    __syncthreads();   // the only cross-wave handoff in the kernel

    if (tid == 0) {
        double T = 0.0, TS = 0.0;
        for (int i = 0; i < BLOCK; ++i) { T += dredT[i]; TS += dredS[i]; }
        double Ure = 0.0, Uim = 0.0, U11 = 0.0;
        for (int w = 0; w < WAVES; ++w) {
            Ure += (double)(wred[w][0] + wred[w][3]);
            Uim += (double)(wred[w][1] + wred[w][4]);
            U11 += (double)(wred[w][2] + wred[w][5]);
        }
        double* o = partials + 5ll * blockIdx.x;
        o[0] = T; o[1] = TS; o[2] = U11; o[3] = Ure; o[4] = Uim;
    }
}

// Kernel 2: reduce block partials (f64), build rho_out in closed form, finish.
__global__ void finalize_kernel(const double* __restrict__ partials, int nblk,
                                const double* __restrict__ kop,
                                float* __restrict__ out)
{
    if (threadIdx.x != 0 || blockIdx.x != 0) return;
    double T = 0.0, TS = 0.0, U11 = 0.0, Ure = 0.0, Uim = 0.0;
    for (int i = 0; i < nblk; ++i) {
        const double* o = partials + 5ll * i;
        T += o[0]; TS += o[1]; U11 += o[2]; Ure += o[3]; Uim += o[4];
    }
    double sq1mg = kop[3];
    double omg   = sq1mg * sq1mg;          // 1-g
    double rho11 = U11 / T;                // tr(rho)=1 by construction
    double r11 = omg * rho11;              // rho_out components
    double r00 = 1.0 - r11;
    double re  = sq1mg * (Ure / T);
    double im  = sq1mg * (Uim / T);
    double det = r00 * r11 - (re * re + im * im);
    double disc = sqrt(fmax(0.0, 1.0 - 4.0 * det));
    double lp = 0.5 * (1.0 + disc), lm = 0.5 * (1.0 - disc);
    double Sout = -(lp * logm_eig_f64(lp) + lm * logm_eig_f64(lm));
    out[0] = (float)(TS / T - Sout);       // -(S(rho_out) - sum c_n S(sigma_n))
}

extern "C" void kernel_launch(void* const* d_in, const int* in_sizes, int n_in,
                              void* d_out, int out_size, void* d_ws, size_t ws_size,
                              hipStream_t stream) {
    const double* theta_coeff = (const double*)d_in[0];
    const double* theta_state = (const double*)d_in[1];
    const double* kop         = (const double*)d_in[2];
    int N = in_sizes[0];

    int grid = 512;                                   // bandwidth-bound: plenty of waves
    size_t need = (size_t)grid * 5 * sizeof(double);
    if (need > ws_size) {
        grid = (int)(ws_size / (5 * sizeof(double)));
        if (grid < 1) grid = 1;
    }
    int total_threads = grid * BLOCK;
    int iters = (N + total_threads - 1) / total_threads;
    double* partials = (double*)d_ws;

    per_state_kernel<<<grid, BLOCK, 0, stream>>>(theta_coeff, theta_state, kop,
                                                 partials, N, total_threads, iters);
    finalize_kernel<<<1, 1, 0, stream>>>(partials, grid, kop, (float*)d_out);
}